// CobraBlock_61692910240067
// MI455X (gfx1250) — compile-verified
//
#include <hip/hip_runtime.h>
#include <math.h>

typedef float v2f __attribute__((ext_vector_type(2)));
typedef float v8f __attribute__((ext_vector_type(8)));

#define DIM     256
#define DT_RANK 16
#define D_STATE 16
#define BATCH   4
#define SEQLEN  4096
#define ROWS    (BATCH * SEQLEN)          // 16384
#define DBC_N   (DT_RANK + 2 * D_STATE)   // 48

// ---------------------------------------------------------------------------
// f32 WMMA GEMM:  C[M x N] = A[M x K] (row stride lda) * B[K x N] (+bias)(+act)
// One wave computes a 16 x (NT*16) strip; 8 waves/block => 128 rows per block.
// B slab for the block's columns is staged in LDS as (K/2) x W float2
// fragment pairs so the k-loop reads are ds_load_b64 with immediate offsets.
// ACT: 0 = none, 1 = softplus
// ---------------------------------------------------------------------------
template<int NT, int K, int ACT>
__global__ __launch_bounds__(256) void gemm_wmma_f32(
    const float* __restrict__ A, int lda,
    const float* __restrict__ B, int N,
    const float* __restrict__ bias,
    float* __restrict__ C)
{
    constexpr int W = NT * 16;                 // columns per block
    __shared__ v2f ldsB[(K / 2) * W];          // <= 64 KB (K=256, W=64)

    const int tid  = threadIdx.x;
    const int wave = tid >> 5;                 // 0..7
    const int lane = tid & 31;
    const int half = lane >> 4;                // 0: K pair 0 / M=r ; 1: pair 1 / M=r+8
    const int lr   = lane & 15;
    const int m0   = (blockIdx.x * 8 + wave) * 16;
    const int n0   = blockIdx.y * W;

    // ---- stage B fragments: ldsB[kk*W + col] = (B[2kk][n0+col], B[2kk+1][n0+col])
    for (int i = tid; i < (K / 2) * W; i += 256) {
        const int kk  = i / W;
        const int col = i - kk * W;
        v2f v;
        v.x = B[(2 * kk)     * N + n0 + col];
        v.y = B[(2 * kk + 1) * N + n0 + col];
        ldsB[i] = v;
    }
    __syncthreads();

    v8f acc[NT] = {};

    const float* Arow  = A + (size_t)(m0 + lr) * lda;
    const v2f*   Bfrag = &ldsB[half * W + lr]; // row (2kk+half), lane's column
#pragma unroll 8
    for (int kk = 0; kk < K / 4; ++kk) {       // k = 4*kk
        const v2f a = *(const v2f*)(Arow + 4 * kk + 2 * half);
#pragma unroll
        for (int t = 0; t < NT; ++t) {
            const v2f b = Bfrag[2 * kk * W + t * 16];
            acc[t] = __builtin_amdgcn_wmma_f32_16x16x4_f32(
                false, a, false, b, (short)0, acc[t], false, false);
        }
    }

#pragma unroll
    for (int t = 0; t < NT; ++t) {
        const int col = n0 + t * 16 + lr;
        const float bv = bias ? bias[col] : 0.0f;
#pragma unroll
        for (int r = 0; r < 8; ++r) {
            const int row = m0 + r + 8 * half;
            float v = acc[t][r] + bv;
            if (ACT == 1) v = (v > 20.0f) ? v : log1pf(__expf(v));
            C[(size_t)row * N + col] = v;
        }
    }
}

// ---------------------------------------------------------------------------
// Depthwise conv1d (k=3, pad=1, per-channel) over sequence axis + SiLU
// ---------------------------------------------------------------------------
__global__ __launch_bounds__(256) void conv_silu_kernel(
    const float* __restrict__ U,
    const float* __restrict__ conv_w,   // (DIM, 3)
    const float* __restrict__ conv_b,   // (DIM,)
    float* __restrict__ xone)
{
    const int idx = blockIdx.x * blockDim.x + threadIdx.x;
    if (idx >= ROWS * DIM) return;
    const int d = idx & (DIM - 1);
    const int l = (idx / DIM) & (SEQLEN - 1);
    const float prev = (l > 0)          ? U[idx - DIM] : 0.0f;
    const float cur  = U[idx];
    const float nxt  = (l < SEQLEN - 1) ? U[idx + DIM] : 0.0f;
    float v = prev * conv_w[d * 3 + 0] + cur * conv_w[d * 3 + 1]
            + nxt  * conv_w[d * 3 + 2] + conv_b[d];
    xone[idx] = v / (1.0f + __expf(-v));   // silu
}

// ---------------------------------------------------------------------------
// Selective scan, fused with D-skip, SiLU gate (on U = x_two) and residual.
// One lane per (b, d, n); 16-lane butterfly reduction over N; 2 channels/wave.
// ---------------------------------------------------------------------------
__global__ __launch_bounds__(256) void scan_kernel(
    const float* __restrict__ delta,    // (ROWS, DIM)
    const float* __restrict__ xone,     // (ROWS, DIM)
    const float* __restrict__ dbc,      // (ROWS, 48): [dt_lr | B | C]
    const float* __restrict__ U,        // (ROWS, DIM)  == x_two
    const float* __restrict__ x,        // (ROWS, DIM)  skip
    const float* __restrict__ A_log,    // (DIM, 16)
    const float* __restrict__ D_skip,   // (DIM,)
    float* __restrict__ outp)           // (ROWS, DIM)
{
    const int wid  = (blockIdx.x * blockDim.x + threadIdx.x) >> 5;
    const int lane = threadIdx.x & 31;
    const int half = lane >> 4;
    const int n    = lane & 15;
    const int b    = wid / (DIM / 2);
    const int d    = (wid % (DIM / 2)) * 2 + half;

    const float A_dn = -__expf(A_log[d * D_STATE + n]);
    const float Dv   = D_skip[d];
    float h = 0.0f;

    const size_t base = (size_t)b * SEQLEN;
    for (int l = 0; l < SEQLEN; ++l) {
        const size_t row = base + l;
        const float dv = delta[row * DIM + d];
        const float xv = xone[row * DIM + d];
        const float Bv = dbc[row * DBC_N + DT_RANK + n];
        const float Cv = dbc[row * DBC_N + DT_RANK + D_STATE + n];

        const float a = __expf(dv * A_dn);
        h = fmaf(a, h, dv * Bv * xv);

        float c = h * Cv;
        c += __shfl_xor(c, 1, 32);
        c += __shfl_xor(c, 2, 32);
        c += __shfl_xor(c, 4, 32);
        c += __shfl_xor(c, 8, 32);

        if (n == 0) {
            const float y  = c + Dv * xv;
            const float g  = U[row * DIM + d];
            const float sg = g / (1.0f + __expf(-g));
            outp[row * DIM + d] = y * sg + x[row * DIM + d];
        }
    }
}

// ---------------------------------------------------------------------------
extern "C" void kernel_launch(void* const* d_in, const int* in_sizes, int n_in,
                              void* d_out, int out_size, void* d_ws, size_t ws_size,
                              hipStream_t stream) {
    const float* x      = (const float*)d_in[0];
    const float* W_proj = (const float*)d_in[1];
    const float* b_proj = (const float*)d_in[2];
    const float* conv_w = (const float*)d_in[3];
    const float* conv_b = (const float*)d_in[4];
    const float* W_dbc  = (const float*)d_in[5];
    const float* W_dt   = (const float*)d_in[6];
    const float* b_dt   = (const float*)d_in[7];
    const float* A_log  = (const float*)d_in[8];
    const float* D_skip = (const float*)d_in[9];
    float* out = (float*)d_out;

    float* ws    = (float*)d_ws;
    float* U     = ws;                           // 16384*256
    float* xone  = U     + (size_t)ROWS * DIM;   // 16384*256
    float* dbc   = xone  + (size_t)ROWS * DIM;   // 16384*48
    float* delta = dbc   + (size_t)ROWS * DBC_N; // 16384*256
    float* outp  = delta + (size_t)ROWS * DIM;   // 16384*256

    const dim3 blk(256);

    // 1) U = x @ W_proj + b_proj   (serves as both conv input and x_two gate)
    gemm_wmma_f32<4, DIM, 0><<<dim3(ROWS / 128, DIM / 64), blk, 0, stream>>>(
        x, DIM, W_proj, DIM, b_proj, U);

    // 2) x_one = silu(depthwise_conv(U))
    conv_silu_kernel<<<(ROWS * DIM + 255) / 256, blk, 0, stream>>>(
        U, conv_w, conv_b, xone);

    // 3) dbc = x_one @ W_dbc                    (N = 48)
    gemm_wmma_f32<3, DIM, 0><<<dim3(ROWS / 128, 1), blk, 0, stream>>>(
        xone, DIM, W_dbc, DBC_N, nullptr, dbc);

    // 4) delta = softplus(dbc[:, :16] @ W_dt + b_dt)   (K = 16, lda = 48)
    gemm_wmma_f32<4, DT_RANK, 1><<<dim3(ROWS / 128, DIM / 64), blk, 0, stream>>>(
        dbc, DBC_N, W_dt, DIM, b_dt, delta);

    // 5) selective scan + D-skip + silu gate + residual
    scan_kernel<<<dim3((BATCH * (DIM / 2) * 32) / 256), blk, 0, stream>>>(
        delta, xone, dbc, U, x, A_log, D_skip, outp);

    // 6) out = outp @ W_proj + b_proj
    gemm_wmma_f32<4, DIM, 0><<<dim3(ROWS / 128, DIM / 64), blk, 0, stream>>>(
        outp, DIM, W_proj, DIM, b_proj, out);
}